// TransformerLayer_56186762166826
// MI455X (gfx1250) — compile-verified
//
#include <hip/hip_runtime.h>
#include <hip/hip_bf16.h>
#include <math.h>

typedef __attribute__((ext_vector_type(2))) float v2f;
typedef __attribute__((ext_vector_type(8))) float v8f;
typedef __attribute__((ext_vector_type(4))) unsigned int v4u;
typedef __attribute__((ext_vector_type(4))) int v4i;
typedef __attribute__((ext_vector_type(8))) int v8i;

#define B_  16
#define S_  1024
#define H_  64
#define NH  4
#define DH  16

#if __has_builtin(__builtin_amdgcn_tensor_load_to_lds) && __has_builtin(__builtin_amdgcn_s_wait_tensorcnt)
#define HAVE_TDM 1
#else
#define HAVE_TDM 0
#endif

__device__ __forceinline__ v8f wmma_f32(v2f a, v2f b, v8f c) {
    // V_WMMA_F32_16X16X4_F32 : D = A(16x4) * B(4x16) + C(16x16), fp32 throughout
    return __builtin_amdgcn_wmma_f32_16x16x4_f32(false, a, false, b, (short)0, c, false, false);
}

#if HAVE_TDM
// Issue a TDM DMA of a [rows x 16] fp32 tile (contiguous, stride 16) into LDS.
__device__ __forceinline__ void tdm_load_v_tile(const float* gsrc, const float* lds_dst) {
    const unsigned long long ga = (unsigned long long)(uintptr_t)gsrc;
    const unsigned int       la = (unsigned int)(uintptr_t)lds_dst; // flat low 32 bits = LDS offset
    v4u g0;
    g0[0] = 1u;                                     // count=1, no gather
    g0[1] = la;                                     // lds_addr [63:32]
    g0[2] = (unsigned int)ga;                       // global_addr [95:64]
    g0[3] = (unsigned int)((ga >> 32) & 0x1FFFFFFu) // global_addr [120:96]
          | 0x80000000u;                            // type=2 ("image") [127:126]
    v8i g1;
    g1[0] = 0x00020000;          // workgroup_mask=0, data_size=2 (4B)
    g1[1] = (int)(16u << 16);    // tensor_dim0 = 16 (bits 79:48, low half)
    g1[2] = (int)(1024u << 16);  // tensor_dim0 hi=0, tensor_dim1 = 1024 (bits 111:80)
    g1[3] = (int)(16u << 16);    // tensor_dim1 hi=0, tile_dim0 = 16
    g1[4] = 1024;                // tile_dim1 = 1024, tile_dim2 = 0
    g1[5] = 16;                  // tensor_dim0_stride = 16
    g1[6] = 0;                   // stride hi / dim1_stride lo
    g1[7] = 0;                   // dim1_stride hi
    v4i z4 = {0, 0, 0, 0};
#if __clang_major__ >= 23
    v8i z8 = {0, 0, 0, 0, 0, 0, 0, 0};
    __builtin_amdgcn_tensor_load_to_lds(g0, g1, z4, z4, z8, 0);
#else
    __builtin_amdgcn_tensor_load_to_lds(g0, g1, z4, z4, 0);
#endif
}
#endif

// ---------------------------------------------------------------------------
// Kernel 1: QKV projection.  X[16384,64] @ W^T + b for Wq/Wk/Wv.
// Block = 128 threads (4 waves). Wave w owns output columns [16w,16w+16) = head w.
// Output layout: [B, NH, S, DH] (head-major) for the attention kernel.
// ---------------------------------------------------------------------------
__global__ __launch_bounds__(128) void k_proj(
    const float* __restrict__ x,
    const float* __restrict__ Wq, const float* __restrict__ bq,
    const float* __restrict__ Wk, const float* __restrict__ bk,
    const float* __restrict__ Wv, const float* __restrict__ bv,
    float* __restrict__ qws, float* __restrict__ kws, float* __restrict__ vws)
{
    const int tid  = threadIdx.x;
    const int wave = tid >> 5;
    const int lane = tid & 31;
    const int row0 = blockIdx.x * 16;      // tile of 16 rows of [B*S, 64]
    const int noff = wave * 16;            // head index == wave
    const int m    = lane & 15;            // A: M index / B: N index
    const int kh   = (lane >> 4) * 2;      // K-half select

    const float* xr = x  + (size_t)(row0 + m) * H_;
    const float* wq = Wq + (size_t)(noff + m) * H_;
    const float* wk = Wk + (size_t)(noff + m) * H_;
    const float* wv = Wv + (size_t)(noff + m) * H_;

    v8f cq = {}; v8f ck = {}; v8f cv = {};
    for (int k0 = 0; k0 < H_; k0 += 4) {
        const int kk = k0 + kh;
        v2f a  = *(const v2f*)(xr + kk);
        v2f fq = *(const v2f*)(wq + kk);
        v2f fk = *(const v2f*)(wk + kk);
        v2f fv = *(const v2f*)(wv + kk);
        cq = wmma_f32(a, fq, cq);
        ck = wmma_f32(a, fk, ck);
        cv = wmma_f32(a, fv, cv);
    }

    const int b  = row0 / S_;
    const int s0 = row0 % S_;
    const float biq = bq[noff + m], bik = bk[noff + m], biv = bv[noff + m];
    #pragma unroll
    for (int r = 0; r < 8; ++r) {
        const int mm = r + ((lane >> 4) << 3);            // C row
        const size_t base = ((size_t)(b * NH + wave) * S_ + (s0 + mm)) * DH + m;
        qws[base] = cq[r] + biq;
        kws[base] = ck[r] + bik;
        vws[base] = cv[r] + biv;
    }
}

// ---------------------------------------------------------------------------
// Kernel 2: attention.  One block per (b, 16-query tile); 256 threads = 8 waves.
// Dynamic LDS (CDNA5: 320KB/WG): att 64KB + head-mean 64KB + V tile 64KB + 8KB.
// V tile is DMAed by the Tensor Data Mover, overlapped with QK^T + softmax.
// ---------------------------------------------------------------------------
__global__ __launch_bounds__(256) void k_attn(
    const float* __restrict__ qws, const float* __restrict__ kws,
    const float* __restrict__ vws, const int* __restrict__ lengths,
    float* __restrict__ ows, float* __restrict__ att_out)
{
    extern __shared__ float smem[];
    float* att    = smem;                     // [16][1024] current head probabilities
    float* attsum = smem + 16 * S_;           // [16][1024] mean over heads
    float* vbuf   = smem + 32 * S_;           // [1024][16] V tile for current head
    float* part   = smem + 32 * S_ + S_ * DH; // [8][16][16] att@V partials

    const int tid  = threadIdx.x;
    const int wave = tid >> 5;
    const int lane = tid & 31;
    const int b    = blockIdx.x >> 6;
    const int q0   = (blockIdx.x & 63) * 16;
    const int len  = lengths[b];

    const int mA = lane & 15;
    const int kh = (lane >> 4) * 2;

    for (int i = tid; i < 16 * S_; i += 256) attsum[i] = 0.0f;

    for (int h = 0; h < NH; ++h) {
        const float* qb = qws + (size_t)(b * NH + h) * S_ * DH;
        const float* kb = kws + (size_t)(b * NH + h) * S_ * DH;
        const float* vb = vws + (size_t)(b * NH + h) * S_ * DH;

        // ---- kick off async V[1024x16] -> LDS (overlaps with QK^T + softmax) ----
#if HAVE_TDM
        if (tid < 32) tdm_load_v_tile(vb, vbuf);
#else
        for (int i = tid; i < S_ * DH; i += 256) vbuf[i] = vb[i];
#endif

        // ---- scores = Q K^T / sqrt(H), masked, into LDS ----
        for (int nt = wave; nt < S_ / 16; nt += 8) {
            __builtin_prefetch(kb + (size_t)((nt + 8) * 16 + mA) * DH, 0, 0);
            v8f c = {};
            #pragma unroll
            for (int k0 = 0; k0 < DH; k0 += 4) {
                const int kk = k0 + kh;
                v2f a  = *(const v2f*)(qb + (size_t)(q0 + mA) * DH + kk);
                v2f bf = *(const v2f*)(kb + (size_t)(nt * 16 + mA) * DH + kk);
                c = wmma_f32(a, bf, c);
            }
            #pragma unroll
            for (int r = 0; r < 8; ++r) {
                const int mm   = r + ((lane >> 4) << 3);
                const int qrow = q0 + mm;
                const int kcol = nt * 16 + (lane & 15);
                const bool ok  = (kcol <= qrow) && (kcol < len) && (qrow < len);
                att[mm * S_ + kcol] = ok ? c[r] * 0.125f : -1e30f;
            }
        }
        __syncthreads();

        // ---- softmax: wave handles rows 2w, 2w+1; lane-strided over 1024 ----
        #pragma unroll
        for (int rr = 0; rr < 2; ++rr) {
            const int row = wave * 2 + rr;
            float* arow = att + row * S_;
            float mx = -3.0e38f;
            for (int i = lane; i < S_; i += 32) mx = fmaxf(mx, arow[i]);
            #pragma unroll
            for (int off = 16; off; off >>= 1) mx = fmaxf(mx, __shfl_xor(mx, off, 32));
            if (mx <= -1e29f) {                 // fully masked row -> zeros
                for (int i = lane; i < S_; i += 32) arow[i] = 0.0f;
            } else {
                float sum = 0.0f;
                for (int i = lane; i < S_; i += 32) {
                    float p = __expf(arow[i] - mx);   // masked: exp(-inf)=0
                    arow[i] = p; sum += p;
                }
                #pragma unroll
                for (int off = 16; off; off >>= 1) sum += __shfl_xor(sum, off, 32);
                const float inv = 1.0f / sum;
                for (int i = lane; i < S_; i += 32) {
                    float p = arow[i] * inv;
                    arow[i] = p;
                    attsum[row * S_ + i] += 0.25f * p;   // mean over 4 heads
                }
            }
        }
        // ---- V tile must be resident before att@V reads it from LDS ----
#if HAVE_TDM
        if (tid < 32) __builtin_amdgcn_s_wait_tensorcnt(0);
#endif
        __syncthreads();

        // ---- out tile = att[16x1024] @ V[1024x16]; K split across 8 waves ----
        // LDS reads: lanes 0-15 hit banks 0-15, lanes 16-31 banks 32-47 (conflict-free)
        v8f c = {};
        for (int k0 = wave * 128; k0 < wave * 128 + 128; k0 += 4) {
            const int kk = k0 + kh;
            v2f a = *(const v2f*)(att + mA * S_ + kk);
            v2f bf;
            bf[0] = vbuf[(kk + 0) * DH + mA];
            bf[1] = vbuf[(kk + 1) * DH + mA];
            c = wmma_f32(a, bf, c);
        }
        #pragma unroll
        for (int r = 0; r < 8; ++r) {
            const int mm = r + ((lane >> 4) << 3);
            part[wave * 256 + mm * 16 + (lane & 15)] = c[r];
        }
        __syncthreads();
        {   // tree reduce 8 partials; each thread owns one of 256 elements
            float s = 0.0f;
            #pragma unroll
            for (int w = 0; w < 8; ++w) s += part[w * 256 + tid];
            const int mm = tid >> 4, nn = tid & 15;
            ows[(size_t)(b * S_ + q0 + mm) * H_ + h * DH + nn] = s;
        }
        __syncthreads();   // protects att/vbuf reuse next head
    }

    // ---- stream head-mean attention to global (write-once: NT hint) ----
    for (int i = tid; i < 16 * S_; i += 256) {
        const int mm = i >> 10, kcol = i & (S_ - 1);
        __builtin_nontemporal_store(attsum[i],
            att_out + (size_t)(b * S_ + q0 + mm) * S_ + kcol);
    }
}

// ---------------------------------------------------------------------------
// Kernel 3: residual + LN1 + FFN(gelu exact) + LN2. 128 threads = 4 waves.
// ---------------------------------------------------------------------------
__global__ __launch_bounds__(128) void k_post(
    const float* __restrict__ x,  const float* __restrict__ ows,
    const float* __restrict__ W1, const float* __restrict__ b1,
    const float* __restrict__ W2, const float* __restrict__ b2,
    const float* __restrict__ g1, const float* __restrict__ be1,
    const float* __restrict__ g2, const float* __restrict__ be2,
    float* __restrict__ y)
{
    __shared__ float t0[16 * H_];   // out1 -> ln1 (in place)
    __shared__ float t1[16 * H_];   // gelu(h1) -> pre-LN2 residual (in place)

    const int tid  = threadIdx.x;
    const int wave = tid >> 5;
    const int lane = tid & 31;
    const int row0 = blockIdx.x * 16;
    const int noff = wave * 16;
    const int mA   = lane & 15;
    const int kh   = (lane >> 4) * 2;

    for (int i = tid; i < 16 * H_; i += 128)
        t0[i] = x[(size_t)row0 * H_ + i] + ows[(size_t)row0 * H_ + i];
    __syncthreads();

    if (tid < 16) {                       // LN1 per row
        float mu = 0.f;
        for (int j = 0; j < H_; ++j) mu += t0[tid * H_ + j];
        mu *= (1.0f / H_);
        float var = 0.f;
        for (int j = 0; j < H_; ++j) { float d = t0[tid * H_ + j] - mu; var += d * d; }
        var *= (1.0f / H_);
        const float rs = rsqrtf(var + 1e-5f);
        for (int j = 0; j < H_; ++j)
            t0[tid * H_ + j] = (t0[tid * H_ + j] - mu) * rs * g1[j] + be1[j];
    }
    __syncthreads();

    {   // GEMM1: gelu(ln1 @ W1^T + b1) -> t1
        v8f c = {};
        for (int k0 = 0; k0 < H_; k0 += 4) {
            const int kk = k0 + kh;
            v2f a  = *(const v2f*)(t0 + mA * H_ + kk);
            v2f bf = *(const v2f*)(W1 + (size_t)(noff + mA) * H_ + kk);
            c = wmma_f32(a, bf, c);
        }
        const float bb = b1[noff + mA];
        #pragma unroll
        for (int r = 0; r < 8; ++r) {
            const int mm = r + ((lane >> 4) << 3);
            float v = c[r] + bb;
            v = 0.5f * v * (1.0f + erff(v * 0.70710678118654752f));  // exact gelu
            t1[mm * H_ + noff + mA] = v;
        }
    }
    __syncthreads();

    {   // GEMM2: t1 @ W2^T + b2 + ln1  -> t1 (after all waves done reading t1)
        v8f c = {};
        for (int k0 = 0; k0 < H_; k0 += 4) {
            const int kk = k0 + kh;
            v2f a  = *(const v2f*)(t1 + mA * H_ + kk);
            v2f bf = *(const v2f*)(W2 + (size_t)(noff + mA) * H_ + kk);
            c = wmma_f32(a, bf, c);
        }
        __syncthreads();
        const float bb = b2[noff + mA];
        #pragma unroll
        for (int r = 0; r < 8; ++r) {
            const int mm = r + ((lane >> 4) << 3);
            t1[mm * H_ + noff + mA] = c[r] + bb + t0[mm * H_ + noff + mA];
        }
    }
    __syncthreads();

    if (tid < 16) {                       // LN2 per row -> y
        float mu = 0.f;
        for (int j = 0; j < H_; ++j) mu += t1[tid * H_ + j];
        mu *= (1.0f / H_);
        float var = 0.f;
        for (int j = 0; j < H_; ++j) { float d = t1[tid * H_ + j] - mu; var += d * d; }
        var *= (1.0f / H_);
        const float rs = rsqrtf(var + 1e-5f);
        for (int j = 0; j < H_; ++j)
            y[(size_t)(row0 + tid) * H_ + j] =
                (t1[tid * H_ + j] - mu) * rs * g2[j] + be2[j];
    }
}

// ---------------------------------------------------------------------------
extern "C" void kernel_launch(void* const* d_in, const int* in_sizes, int n_in,
                              void* d_out, int out_size, void* d_ws, size_t ws_size,
                              hipStream_t stream)
{
    (void)in_sizes; (void)n_in; (void)out_size; (void)ws_size;
    const float* x   = (const float*)d_in[0];
    const int*   len = (const int*)  d_in[1];
    // d_in[2] = atten (unused placeholder)
    const float* Wq = (const float*)d_in[3];  const float* bq = (const float*)d_in[4];
    const float* Wk = (const float*)d_in[5];  const float* bk = (const float*)d_in[6];
    const float* Wv = (const float*)d_in[7];  const float* bv = (const float*)d_in[8];
    const float* W1 = (const float*)d_in[9];  const float* b1 = (const float*)d_in[10];
    const float* W2 = (const float*)d_in[11]; const float* b2 = (const float*)d_in[12];
    const float* g1 = (const float*)d_in[13]; const float* be1= (const float*)d_in[14];
    const float* g2 = (const float*)d_in[15]; const float* be2= (const float*)d_in[16];

    float* out     = (float*)d_out;                     // y [B,S,H] then att_mean [B,S,S]
    float* att_out = out + (size_t)B_ * S_ * H_;

    const size_t qkv = (size_t)B_ * NH * S_ * DH;       // 1M floats each
    float* qws = (float*)d_ws;
    float* kws = qws + qkv;
    float* vws = kws + qkv;
    float* ows = vws + qkv;

    const int tiles = B_ * S_ / 16;                     // 1024
    k_proj<<<tiles, 128, 0, stream>>>(x, Wq, bq, Wk, bk, Wv, bv, qws, kws, vws);

    // att 64KB + head-mean 64KB + V tile 64KB + partials 8KB = 200KB (<320KB/WG on CDNA5)
    const size_t smem = (size_t)(32 * S_ + S_ * DH + 8 * 256) * sizeof(float);
    k_attn<<<B_ * (S_ / 16), 256, smem, stream>>>(qws, kws, vws, len, ows, att_out);

    k_post<<<tiles, 128, 0, stream>>>(x, ows, W1, b1, W2, b2, g1, be1, g2, be2, out);
}